// DLRM_4166118277508
// MI455X (gfx1250) — compile-verified
//
#include <hip/hip_runtime.h>

// Per-batch Gram matrix + strict-lower-tri extraction for MI455X (gfx1250).
// B=8192 samples, each X_b is 64x128 fp32. Z_b = X_b * X_b^T (64x64),
// output = Z_b[i,j] for j < i, i-major -> 2016 floats per sample.
//
// Memory-bound problem (~322 MB traffic -> ~14us at 23.3 TB/s), so we keep
// fp32 end-to-end and use V_WMMA_F32_16X16X4_F32 for the tile products.

typedef __attribute__((ext_vector_type(2))) float v2f;
typedef __attribute__((ext_vector_type(8))) float v8f;

#define NROWS 64
#define DDIM  128
// Padded LDS row stride (floats). 130 floats = 520 bytes: row r starts at
// DWORD-bank (2*r) % 64, so 16 rows fetched by lanes 0..15 land on 16
// distinct bank pairs for ds_load_b64 (stride 128 would be a 16-way conflict).
#define LDSTRIDE 130
#define OUT_PER_B 2016  // 64*63/2

__global__ __launch_bounds__(320) void gram_lowtri_wmma(
    const float* __restrict__ X, float* __restrict__ out) {
  __shared__ float lds[NROWS * LDSTRIDE];

  const int b   = blockIdx.x;
  const int tid = threadIdx.x;
  const float* __restrict__ src = X + (size_t)b * (NROWS * DDIM);

  // ---- Stage X_b (64x128 f32 = 32KB) into padded LDS ----
  // 2048 float4 global loads spread over 320 threads; split each into two
  // ds_store_b64 because the padded stride keeps only 8-byte alignment.
  for (int idx = tid; idx < (NROWS * DDIM) / 4; idx += 320) {
    const int row = idx >> 5;         // 32 float4 per row
    const int c4  = (idx & 31) << 2;  // float offset within the row
    const float4 v = *(const float4*)(src + row * DDIM + c4);
    v2f lo; lo.x = v.x; lo.y = v.y;
    v2f hi; hi.x = v.z; hi.y = v.w;
    *(v2f*)(&lds[row * LDSTRIDE + c4])     = lo;
    *(v2f*)(&lds[row * LDSTRIDE + c4 + 2]) = hi;
  }
  __syncthreads();

  // ---- Map wave -> lower-triangular 16x16 tile (ti >= tj) ----
  // w: 0..9 -> (0,0),(1,0),(1,1),(2,0),(2,1),(2,2),(3,0),(3,1),(3,2),(3,3)
  const int w = tid >> 5;
  int ti = 0, base = 0;
  while (base + ti + 1 <= w) { base += ti + 1; ++ti; }
  const int tj = w - base;

  const int lane = tid & 31;
  const int nlo  = lane & 15;  // row-within-strip for A/B fetch; N for C/D
  const int half = lane >> 4;  // selects K=0,1 vs K=2,3 halves

  // A (16x4 f32) lane layout: lanes 0-15 hold {K=0,K=1}, lanes 16-31 {K=2,K=3}
  // of row M=lane%16. B (4x16) of the Gram product is the transposed slice of
  // another row strip of X, whose register image is the mirrored pattern, so
  // both operands use the identical ds_load_b64 addressing.
  const float* __restrict__ Ar = &lds[(ti * 16 + nlo) * LDSTRIDE + 2 * half];
  const float* __restrict__ Br = &lds[(tj * 16 + nlo) * LDSTRIDE + 2 * half];

  v8f c = {0.f, 0.f, 0.f, 0.f, 0.f, 0.f, 0.f, 0.f};
#pragma unroll
  for (int k = 0; k < DDIM; k += 4) {
    const v2f a  = *(const v2f*)(Ar + k);
    const v2f bm = *(const v2f*)(Br + k);
    // D = A(16x4) x B(4x16) + C, fp32 accumulate
    c = __builtin_amdgcn_wmma_f32_16x16x4_f32(
        /*neg_a=*/false, a, /*neg_b=*/false, bm,
        /*c_mod=*/(short)0, c, /*reuse_a=*/false, /*reuse_b=*/false);
  }

  // ---- Scatter strict-lower-tri entries ----
  // C/D layout: VGPR r, lanes 0-15 -> (M=r, N=lane); lanes 16-31 -> (M=r+8).
  // Within a row i the 16 lanes hit consecutive addresses -> coalesced.
  float* __restrict__ ob = out + (size_t)b * OUT_PER_B;
#pragma unroll
  for (int r = 0; r < 8; ++r) {
    const int m = r + 8 * half;
    const int i = 16 * ti + m;
    const int j = 16 * tj + nlo;
    if (j < i) {  // always true for off-diagonal tiles; prunes diagonal ones
      ob[((i * (i - 1)) >> 1) + j] = c[r];
    }
  }
}

extern "C" void kernel_launch(void* const* d_in, const int* in_sizes, int n_in,
                              void* d_out, int out_size, void* d_ws, size_t ws_size,
                              hipStream_t stream) {
  (void)n_in; (void)out_size; (void)d_ws; (void)ws_size;
  const float* X = (const float*)d_in[0];
  float* out = (float*)d_out;
  const int B = in_sizes[0] / (NROWS * DDIM);  // 8192 for the reference shape
  gram_lowtri_wmma<<<B, 320, 0, stream>>>(X, out);
}